// RelationalGATLayer_22840636080684
// MI455X (gfx1250) — compile-verified
//
#include <hip/hip_runtime.h>
#include <hip/hip_bf16.h>

#define NN 50000      // N_NODES
#define NE 600000     // N_EDGES
#define NR 500        // N_RELS
#define DIM 128
#define MT 4          // row tiles (of 16) per wave in the WMMA GEMM

typedef __attribute__((ext_vector_type(2))) float v2f;
typedef __attribute__((ext_vector_type(8))) float v8f;

// ---------------------------------------------------------------------------
// out[i][j] = sum_k A[i][k] * W[j][k] + bias[j]      (i.e.  A @ W^T + b)
// One wave32 per (64-row x 16-col) strip: 4 accumulator tiles share every
// B (weight) load, K swept 4 at a time with V_WMMA_F32_16X16X4_F32.
//   A-tile 16x4 f32: lane = M (lane&15), lane-half selects K pair -> float2
//   B-tile 4x16 f32: lane = N (lane&15), lane-half selects K pair -> float2
//   -> B[k][n] = W[n][k]: contiguous float2 per lane, no transpose needed.
// C/D: VGPR v holds (M = (lane>>4)*8 + v, N = lane&15).
// Tiles are 16 rows; callers have nrows s.t. each tile is fully valid or
// fully out-of-range except possibly the last (rel GEMM), handled raggedly.
// ---------------------------------------------------------------------------
__global__ __launch_bounds__(32) void gemm_wmma_f32(
    const float* __restrict__ A, const float* __restrict__ W,
    const float* __restrict__ bias, float* __restrict__ out, int nrows) {
  const int lane = threadIdx.x & 31;
  const int half = lane >> 4;    // 0 for lanes 0-15, 1 for lanes 16-31
  const int l15  = lane & 15;
  const int rbase = blockIdx.x * (16 * MT);
  const int cbase = blockIdx.y * 16;

  const float* __restrict__ Wrow =
      W + (size_t)(cbase + l15) * DIM + 2 * half;

  const float* Arow[MT];
#pragma unroll
  for (int t = 0; t < MT; ++t) {
    int r = rbase + 16 * t + l15;
    if (r >= nrows) r = nrows - 1;               // clamp loads; stores guarded
    Arow[t] = A + (size_t)r * DIM + 2 * half;
  }

  v8f acc[MT];
#pragma unroll
  for (int t = 0; t < MT; ++t) acc[t] = (v8f){};

#pragma unroll 4
  for (int k0 = 0; k0 < DIM; k0 += 4) {
    const v2f b = *(const v2f*)(Wrow + k0);      // shared across all row tiles
#pragma unroll
    for (int t = 0; t < MT; ++t) {
      const v2f a = *(const v2f*)(Arow[t] + k0);
      acc[t] = __builtin_amdgcn_wmma_f32_16x16x4_f32(
          /*neg_a=*/false, a, /*neg_b=*/false, b,
          /*c_mod=*/(short)0, acc[t], /*reuse_a=*/false, /*reuse_b=*/false);
    }
  }

  const float bv = bias[cbase + l15];
#pragma unroll
  for (int t = 0; t < MT; ++t) {
    const int tb = rbase + 16 * t;
    if (tb + 16 <= nrows) {                      // uniform branch: full tile
#pragma unroll
      for (int v = 0; v < 8; ++v)
        out[(size_t)(tb + half * 8 + v) * DIM + cbase + l15] = acc[t][v] + bv;
    } else if (tb < nrows) {                     // ragged tile (rel GEMM only)
#pragma unroll
      for (int v = 0; v < 8; ++v) {
        const int row = tb + half * 8 + v;
        if (row < nrows)
          out[(size_t)row * DIM + cbase + l15] = acc[t][v] + bv;
      }
    }
  }
}

// ---------------------------------------------------------------------------
// Fused pair of row dots: o1[r] = X[r]·w1, o2[r] = X[r]·w2 (one wave per row,
// X row read once).
// ---------------------------------------------------------------------------
__global__ void rowdot2_kernel(const float* __restrict__ X,
                               const float* __restrict__ w1,
                               const float* __restrict__ w2,
                               float* __restrict__ o1, float* __restrict__ o2,
                               int nrows) {
  const int wid  = (blockIdx.x * blockDim.x + threadIdx.x) >> 5;
  const int lane = threadIdx.x & 31;
  if (wid >= nrows) return;
  const float* row = X + (size_t)wid * DIM;
  float a1 = 0.f, a2 = 0.f;
#pragma unroll
  for (int t = 0; t < 4; ++t) {
    const float x = row[lane + 32 * t];
    a1 += x * w1[lane + 32 * t];
    a2 += x * w2[lane + 32 * t];
  }
#pragma unroll
  for (int off = 16; off > 0; off >>= 1) {
    a1 += __shfl_xor(a1, off, 32);
    a2 += __shfl_xor(a2, off, 32);
  }
  if (lane == 0) { o1[wid] = a1; o2[wid] = a2; }
}

__global__ void rowdot_kernel(const float* __restrict__ X,
                              const float* __restrict__ w,
                              float* __restrict__ out, int nrows) {
  const int wid  = (blockIdx.x * blockDim.x + threadIdx.x) >> 5;
  const int lane = threadIdx.x & 31;
  if (wid >= nrows) return;
  const float* row = X + (size_t)wid * DIM;
  float acc = 0.f;
#pragma unroll
  for (int t = 0; t < 4; ++t) acc += row[lane + 32 * t] * w[lane + 32 * t];
#pragma unroll
  for (int off = 16; off > 0; off >>= 1) acc += __shfl_xor(acc, off, 32);
  if (lane == 0) out[wid] = acc;
}

// ---------------------------------------------------------------------------
__global__ void init_kernel(float* __restrict__ out, float* __restrict__ segmax,
                            float* __restrict__ denom) {
  const int idx = blockIdx.x * blockDim.x + threadIdx.x;
  if (idx < NN * DIM) out[idx] = 0.f;
  if (idx < NN) {
    segmax[idx] = __int_as_float((int)0xff800000u);  // -inf
    denom[idx]  = 0.f;
  }
}

__device__ __forceinline__ void atomicMaxF32(float* addr, float v) {
  // Correct for mixed signs given -inf initialization.
  if (v >= 0.f) atomicMax((int*)addr, __float_as_int(v));
  else          atomicMin((unsigned int*)addr, __float_as_uint(v));
}

__global__ void edge_scores_kernel(const float* __restrict__ s_src,
                                   const float* __restrict__ s_tgt,
                                   const float* __restrict__ s_rel,
                                   const int* __restrict__ src,
                                   const int* __restrict__ dst,
                                   const int* __restrict__ rel,
                                   const float* __restrict__ attn_b,
                                   float* __restrict__ scores,
                                   float* __restrict__ segmax) {
  const int e = blockIdx.x * blockDim.x + threadIdx.x;
  if (e >= NE) return;
  const float sc = s_src[src[e]] + s_tgt[dst[e]] + s_rel[rel[e]] + attn_b[0];
  scores[e] = sc;
  atomicMaxF32(segmax + dst[e], sc);
}

__global__ void edge_exp_kernel(const float* __restrict__ scores,
                                const int* __restrict__ dst,
                                const float* __restrict__ segmax,
                                float* __restrict__ ex,
                                float* __restrict__ denom) {
  const int e = blockIdx.x * blockDim.x + threadIdx.x;
  if (e >= NE) return;
  const int d = dst[e];
  const float v = expf(scores[e] - segmax[d]);
  ex[e] = v;
  atomicAdd(denom + d, v);
}

// One 128-thread block per edge: coalesced gather + coalesced f32 atomics
// into the (L2-resident) output accumulator.
__global__ __launch_bounds__(DIM) void edge_scatter_kernel(
    const float* __restrict__ ex, const float* __restrict__ denom,
    const int* __restrict__ src, const int* __restrict__ dst,
    const int* __restrict__ rel, const float* __restrict__ m_node,
    const float* __restrict__ m_rel, float* __restrict__ out) {
  const int e = blockIdx.x;
  const int j = threadIdx.x;
  const int d = dst[e];
  const float alpha = ex[e] / denom[d];
  const float msg = m_node[(size_t)src[e] * DIM + j] +
                    m_rel[(size_t)rel[e] * DIM + j];
  atomicAdd(out + (size_t)d * DIM + j, alpha * msg);
}

__global__ void finalize_kernel(const float* __restrict__ denom,
                                const float* __restrict__ node_emb,
                                float* __restrict__ out) {
  const int idx = blockIdx.x * blockDim.x + threadIdx.x;
  if (idx >= NN * DIM) return;
  const int i = idx >> 7;  // / DIM
  if (!(denom[i] > 0.f)) out[idx] = node_emb[idx];  // deg==0 <=> denom==0
}

// ---------------------------------------------------------------------------
extern "C" void kernel_launch(void* const* d_in, const int* in_sizes, int n_in,
                              void* d_out, int out_size, void* d_ws,
                              size_t ws_size, hipStream_t stream) {
  const float* node_emb = (const float*)d_in[0];
  const float* rel_emb  = (const float*)d_in[1];
  const float* W_node_w = (const float*)d_in[2];
  const float* W_node_b = (const float*)d_in[3];
  const float* W_rel_w  = (const float*)d_in[4];
  const float* W_rel_b  = (const float*)d_in[5];
  const float* attn_w   = (const float*)d_in[6];
  const float* attn_b   = (const float*)d_in[7];
  const int*   src      = (const int*)d_in[8];
  const int*   dst      = (const int*)d_in[9];
  const int*   rel      = (const int*)d_in[10];
  float* out = (float*)d_out;

  // Workspace layout (floats), total ~31.5 MB
  float* w0 = (float*)d_ws;
  float* s_src  = w0;              // 50048
  float* s_tgt  = s_src + 50048;   // 50048
  float* s_rel  = s_tgt + 50048;   // 512
  float* segmax = s_rel + 512;     // 50048
  float* denom  = segmax + 50048;  // 50048
  float* scores = denom + 50048;   // 600064
  float* ex     = scores + 600064; // 600064
  float* m_rel  = ex + 600064;     // 512*128
  float* m_node = m_rel + 512 * DIM; // 50000*128

  // 1) init accumulators / output (d_out not re-poisoned between replays)
  init_kernel<<<(NN * DIM) / 256, 256, 0, stream>>>(out, segmax, denom);

  // 2) attention projections (wave per row; node row read once for both dots)
  rowdot2_kernel<<<(NN + 7) / 8, 256, 0, stream>>>(node_emb, attn_w,
                                                   attn_w + DIM, s_src, s_tgt,
                                                   NN);
  rowdot_kernel<<<(NR + 7) / 8, 256, 0, stream>>>(rel_emb, attn_w + 2 * DIM,
                                                  s_rel, NR);

  // 3) WMMA GEMMs:  m = X @ W^T + b   (64 rows x 16 cols per wave)
  gemm_wmma_f32<<<dim3((NN + 16 * MT - 1) / (16 * MT), DIM / 16), 32, 0,
                  stream>>>(node_emb, W_node_w, W_node_b, m_node, NN);
  gemm_wmma_f32<<<dim3((NR + 16 * MT - 1) / (16 * MT), DIM / 16), 32, 0,
                  stream>>>(rel_emb, W_rel_w, W_rel_b, m_rel, NR);

  // 4) edge scores + segment max
  edge_scores_kernel<<<(NE + 255) / 256, 256, 0, stream>>>(
      s_src, s_tgt, s_rel, src, dst, rel, attn_b, scores, segmax);

  // 5) exp + segment sum (softmax denominator)
  edge_exp_kernel<<<(NE + 255) / 256, 256, 0, stream>>>(scores, dst, segmax,
                                                        ex, denom);

  // 6) weighted message scatter-add
  edge_scatter_kernel<<<NE, DIM, 0, stream>>>(ex, denom, src, dst, rel, m_node,
                                              m_rel, out);

  // 7) fallback for isolated nodes
  finalize_kernel<<<(NN * DIM) / 256, 256, 0, stream>>>(denom, node_emb, out);
}